// SpatialConstraint_3307124818456
// MI455X (gfx1250) — compile-verified
//
#include <hip/hip_runtime.h>
#include <hip/hip_bf16.h>

// ---------------------------------------------------------------------------
// consistency loss: (1/n) * sum_i sum_j w_ij * ||p_i - p_(n_ij)||^2
//   probs: [n,128] f32, weights: [n,32] f32, neighbor_idx: [n,32] i32
// Roofline: ~845 MB of gathered reads vs ~0.6 GFLOP -> L2-gather bound
// (25.6 MB table resident in 192 MB L2). Main loop: wave32 per spot, f32 FMA,
// 4-deep global_load_async_to_lds_b128 pipeline (ASYNCcnt) with 8 LDS row
// buffers per wave. Final reduction: exact f32 v_wmma_f32_16x16x4_f32 row-sum.
// ---------------------------------------------------------------------------

#define WAVES_PER_BLOCK 8
#define BLOCK_THREADS   256
#define NDIM            128
#define KNEIGH          32
#define NBUF            8     // LDS row buffers per wave (power of 2)
#define DEPTH           4     // async loads in flight per wave

typedef __attribute__((ext_vector_type(2))) float v2f;
typedef __attribute__((ext_vector_type(8))) float v8f;

// must match the async builtin's parameter type: 'int __vector(4)'
typedef int b128_t __attribute__((vector_size(16)));
typedef __attribute__((address_space(1))) b128_t* global_b128_ptr;
typedef __attribute__((address_space(3))) b128_t* lds_b128_ptr;

#if __has_builtin(__builtin_amdgcn_global_load_async_to_lds_b128) && \
    __has_builtin(__builtin_amdgcn_s_wait_asynccnt)
#define USE_ASYNC_LDS 1
#else
#define USE_ASYNC_LDS 0
#endif

#if USE_ASYNC_LDS
__device__ __forceinline__ void async_row_load16(const float* gsrc, float* lds_dst) {
  // one wave issue = 32 lanes x 16B = one full 512B neighbor row -> LDS
  __builtin_amdgcn_global_load_async_to_lds_b128(
      (global_b128_ptr)(b128_t*)(void*)gsrc,
      (lds_b128_ptr)(b128_t*)(void*)lds_dst,
      0, 0);
}
#endif

__global__ __launch_bounds__(BLOCK_THREADS)
void spatial_consistency_kernel(const float* __restrict__ probs,
                                const float* __restrict__ wts,
                                const int*   __restrict__ nidx,
                                float* __restrict__ partials,
                                int nspots) {
  __shared__ __align__(16) float lbuf[WAVES_PER_BLOCK][NBUF][NDIM];
  __shared__ float wpart[WAVES_PER_BLOCK];

  const int lane = threadIdx.x & 31;
  const int wv   = threadIdx.x >> 5;
  const int gw      = blockIdx.x * WAVES_PER_BLOCK + wv;
  const int gstride = gridDim.x * WAVES_PER_BLOCK;

  float lane_acc = 0.0f;  // this lane's 4-dim partial, summed over spots & neighbors

  for (int i = gw; i < nspots; i += gstride) {
    // own row: lane owns dims [4*lane, 4*lane+4)  (global_load_b128)
    const float4 p = *(const float4*)(probs + (size_t)i * NDIM + lane * 4);
    // this lane carries neighbor slot `lane`'s index & weight; broadcast per j
    const int   nb  = nidx[(size_t)i * KNEIGH + lane];
    const float wgt = wts [(size_t)i * KNEIGH + lane];

    // prefetch next spot's own row into cache (global_prefetch_b8)
    {
      int inext = i + gstride;
      if (inext < nspots)
        __builtin_prefetch(probs + (size_t)inext * NDIM + lane * 4, 0, 3);
    }

    // accumulate w_ij * ||p_i - p_j||^2 partials for this lane's 4 dims
    auto compute = [&](int j) {
      const float4 q = *(const float4*)&lbuf[wv][j & (NBUF - 1)][lane * 4];
      const float wj = __shfl(wgt, j, 32);
      const float dx = p.x - q.x, dy = p.y - q.y, dz = p.z - q.z, dw = p.w - q.w;
      float t = fmaf(dx, dx, fmaf(dy, dy, fmaf(dz, dz, dw * dw)));
      lane_acc = fmaf(wj, t, lane_acc);
    };

#if USE_ASYNC_LDS
    auto issue = [&](int r) {
      int nr = __shfl(nb, r, 32);
      async_row_load16(probs + (size_t)nr * NDIM + lane * 4,
                       &lbuf[wv][r & (NBUF - 1)][lane * 4]);
    };
    // prologue: stage rows 0..DEPTH-2
    issue(0); issue(1); issue(2);
    // steady state: DEPTH rows in flight; wait asynccnt<=DEPTH-1 => row j done
    for (int j = 0; j <= KNEIGH - DEPTH; ++j) {      // j = 0..28 issues rows 3..31
      issue(j + DEPTH - 1);
      __builtin_amdgcn_s_wait_asynccnt(3);
      compute(j);
    }
    // epilogue: drain (wait counts must be literal immediates)
    __builtin_amdgcn_s_wait_asynccnt(2); compute(KNEIGH - 3);
    __builtin_amdgcn_s_wait_asynccnt(1); compute(KNEIGH - 2);
    __builtin_amdgcn_s_wait_asynccnt(0); compute(KNEIGH - 1);
#else
    for (int j = 0; j < KNEIGH; ++j) {
      int nj = __shfl(nb, j, 32);
      const float4 q = *(const float4*)(probs + (size_t)nj * NDIM + lane * 4);
      const float wj = __shfl(wgt, j, 32);
      const float dx = p.x - q.x, dy = p.y - q.y, dz = p.z - q.z, dw = p.w - q.w;
      float t = fmaf(dx, dx, fmaf(dy, dy, fmaf(dz, dz, dw * dw)));
      lane_acc = fmaf(wj, t, lane_acc);
    }
#endif
  }

  // wave reduce (32 lanes -> scalar)
  float a = lane_acc;
  for (int m = 16; m >= 1; m >>= 1) a += __shfl_xor(a, m, 32);
  if (lane == 0) wpart[wv] = a;
  __syncthreads();
  if (threadIdx.x == 0) {
    float s = 0.0f;
    #pragma unroll
    for (int k = 0; k < WAVES_PER_BLOCK; ++k) s += wpart[k];
    partials[blockIdx.x] = s;  // deterministic: fixed block partition, no atomics
  }
}

// Final reduce: single wave32. WMMA f32 16x16x4 with a ones B-matrix is an
// exact f32 cross-lane sum: A[m][0]=acc_m (lanes 0-15), A[m][2]=acc_{m+16}
// (lanes 16-31), A[*][1]=A[*][3]=0 -> D[m][n] = acc_m + acc_{m+16} for all n.
__global__ __launch_bounds__(32)
void reduce_partials_kernel(const float* __restrict__ parts,
                            float* __restrict__ out,
                            int nparts, float inv_n) {
  const int lane = threadIdx.x;
  float acc = 0.0f;
  for (int t = lane; t < nparts; t += 32) acc += parts[t];

#if __has_builtin(__builtin_amdgcn_wmma_f32_16x16x4_f32)
  v2f av; av[0] = acc;  av[1] = 0.0f;   // A: 16x4 f32 striped across the wave
  v2f bv; bv[0] = 1.0f; bv[1] = 1.0f;   // B: 4x16 all-ones
  v8f cv = {};
  v8f dm = __builtin_amdgcn_wmma_f32_16x16x4_f32(
      /*neg_a=*/false, av, /*neg_b=*/false, bv,
      /*c_mod=*/(short)0, cv, /*reuse_a=*/false, /*reuse_b=*/false);
  float t8 = 0.0f;
  #pragma unroll
  for (int r = 0; r < 8; ++r) t8 += dm[r];   // lanes 0-15: rows 0-7; 16-31: rows 8-15
  float total = __shfl(t8, 0, 32) + __shfl(t8, 16, 32);
#else
  float total = acc;
  for (int m = 16; m >= 1; m >>= 1) total += __shfl_xor(total, m, 32);
#endif

  if (lane == 0) out[0] = total * inv_n;
}

extern "C" void kernel_launch(void* const* d_in, const int* in_sizes, int n_in,
                              void* d_out, int out_size, void* d_ws, size_t ws_size,
                              hipStream_t stream) {
  const float* probs = (const float*)d_in[0];   // [n,128] f32
  const float* wts   = (const float*)d_in[1];   // [n,32]  f32
  const int*   nidx  = (const int*)  d_in[2];   // [n,32]  i32

  const int nspots = in_sizes[2] / KNEIGH;

  int nblocks = 512;                                   // 4096 waves -> ~12 spots/wave
  size_t maxp = ws_size / sizeof(float);
  if ((size_t)nblocks > maxp) nblocks = (int)maxp;
  if (nblocks < 1) nblocks = 1;

  float* partials = (float*)d_ws;

  spatial_consistency_kernel<<<nblocks, BLOCK_THREADS, 0, stream>>>(
      probs, wts, nidx, partials, nspots);
  reduce_partials_kernel<<<1, 32, 0, stream>>>(
      partials, (float*)d_out, nblocks, 1.0f / (float)nspots);
}